// AttnDecoderGRU_30588757082531
// MI455X (gfx1250) — compile-verified
//
#include <hip/hip_runtime.h>
#include <math.h>

#define Bb 128
#define Ss 128
#define Hh 512
#define Vv 32000
#define Tt 5
#define BOS_TOK 1

typedef __attribute__((ext_vector_type(2))) float v2f;
typedef __attribute__((ext_vector_type(8))) float v8f;

// ---------------------------------------------------------------------------
// Fragment loader. K is compile-time so i*16*K*4 byte offsets (<= 192KB) fold
// into the 24-bit global_load immediate; the .x/.y pairs merge into
// global_load_b64. aBase/wBase are per-lane row pointers (+2*lhi K-split).
// ---------------------------------------------------------------------------
template <int K>
__device__ __forceinline__ void load_frags(const float* __restrict__ aBase,
                                           const float* __restrict__ wBase,
                                           int k, v2f fa[4], v2f fb[4]) {
#pragma unroll
  for (int i = 0; i < 4; ++i) {
    const float* p = aBase + k + i * 16 * K;
    fa[i].x = p[0]; fa[i].y = p[1];
  }
#pragma unroll
  for (int j = 0; j < 4; ++j) {
    const float* p = wBase + k + j * 16 * K;
    fb[j].x = p[0]; fb[j].y = p[1];
  }
}

__device__ __forceinline__ void wmma_tile(v8f acc[4][4], const v2f fa[4],
                                          const v2f fb[4]) {
#pragma unroll
  for (int i = 0; i < 4; ++i)
#pragma unroll
    for (int j = 0; j < 4; ++j)
      acc[i][j] = __builtin_amdgcn_wmma_f32_16x16x4_f32(
          false, fa[i], false, fb[j], (short)0, acc[i][j], false, false);
}

// ---------------------------------------------------------------------------
// WMMA f32 GEMM:  C[m,n] = sum_k A[m,k] * W[n,k] + bias[n]
// A: M x K row-major;  W: N x K row-major (B = W^T);  C row stride ldc.
// One wave owns a 64(M) x 64(N) tile: 4 A-frags x 4 B-frags -> 16 WMMA per
// K-step of 4, software-pipelined one K-step ahead, last step peeled so all
// prefetch offsets are affine (no per-iteration selects).
// Fragment layouts per CDNA5 ISA 7.12.2 (f32 16x16x4):
//   A: lane<16 -> (K=k0,k0+1), lane>=16 -> (K=k0+2,k0+3), m = lane&15
//   B: same K split by lane-half, n = lane&15
//   C/D: VGPR v -> m = v + 8*(lane>>4), n = lane&15
// ---------------------------------------------------------------------------
template <int K>
__global__ __launch_bounds__(256) void wmma_gemm_nt(
    const float* __restrict__ A, const float* __restrict__ W,
    const float* __restrict__ bias, float* __restrict__ C,
    int M, int N, int ldc) {
  const int wave = blockIdx.x * (blockDim.x >> 5) + (threadIdx.x >> 5);
  const int lane = threadIdx.x & 31;
  const int lrow = lane & 15;
  const int lhi  = lane >> 4;
  const int tilesN = N >> 6;
  const int tm = wave / tilesN;
  const int tn = wave - tm * tilesN;
  if (tm >= (M >> 6)) return;          // wave-uniform: EXEC all-ones for WMMA
  const int m0 = tm << 6;
  const int n0 = tn << 6;

  v8f acc[4][4];
#pragma unroll
  for (int j = 0; j < 4; ++j) {
    const float bv = bias ? bias[n0 + j * 16 + lrow] : 0.0f;
#pragma unroll
    for (int i = 0; i < 4; ++i)
#pragma unroll
      for (int v = 0; v < 8; ++v) acc[i][j][v] = bv;
  }

  const float* aBase = A + (size_t)(m0 + lrow) * K + 2 * lhi;
  const float* wBase = W + (size_t)(n0 + lrow) * K + 2 * lhi;

  v2f fa[4], fb[4];
  load_frags<K>(aBase, wBase, 0, fa, fb);

#pragma unroll 2
  for (int k = 0; k < K - 4; k += 4) {
    v2f na[4], nb[4];
    load_frags<K>(aBase, wBase, k + 4, na, nb);  // prefetch next K-step
    wmma_tile(acc, fa, fb);
#pragma unroll
    for (int i = 0; i < 4; ++i) { fa[i] = na[i]; fb[i] = nb[i]; }
  }
  wmma_tile(acc, fa, fb);                        // peeled final K-step

#pragma unroll
  for (int i = 0; i < 4; ++i)
#pragma unroll
    for (int j = 0; j < 4; ++j)
#pragma unroll
      for (int v = 0; v < 8; ++v)
        C[(size_t)(m0 + i * 16 + v + 8 * lhi) * ldc + (n0 + j * 16 + lrow)] =
            acc[i][j][v];
}

// ---------------------------------------------------------------------------
// scores[b,s] = va . tanh(q[b,:] + ua[b,s,:]) + Va_b   (one wave per (b,s) row)
// ---------------------------------------------------------------------------
__global__ __launch_bounds__(256) void attn_scores(
    const float* __restrict__ q, const float* __restrict__ ua,
    const float* __restrict__ va, const float* __restrict__ vab,
    float* __restrict__ scores) {
  const int wave = (blockIdx.x * blockDim.x + threadIdx.x) >> 5;
  const int lane = threadIdx.x & 31;
  if (wave >= Bb * Ss) return;
  const int b = wave >> 7;  // / Ss
  const float* qb = q + (size_t)b * Hh;
  const float* ub = ua + (size_t)wave * Hh;
  float sum = 0.0f;
  for (int h = lane; h < Hh; h += 32) sum += va[h] * tanhf(qb[h] + ub[h]);
  for (int o = 16; o > 0; o >>= 1) sum += __shfl_xor(sum, o, 32);
  if (lane == 0) scores[wave] = sum + vab[0];
}

// ---------------------------------------------------------------------------
// softmax over S=128, one wave per batch; writes weights + attention output
// ---------------------------------------------------------------------------
__global__ __launch_bounds__(128) void attn_softmax(
    const float* __restrict__ scores, float* __restrict__ w,
    float* __restrict__ attn_out, int t) {
  const int wave = (blockIdx.x * blockDim.x + threadIdx.x) >> 5;
  const int lane = threadIdx.x & 31;
  if (wave >= Bb) return;
  const float* sr = scores + (size_t)wave * Ss;
  float v[4];
  float m = -3.402823466e38f;
#pragma unroll
  for (int i = 0; i < 4; ++i) { v[i] = sr[lane + 32 * i]; m = fmaxf(m, v[i]); }
  for (int o = 16; o > 0; o >>= 1) m = fmaxf(m, __shfl_xor(m, o, 32));
  float s = 0.0f;
#pragma unroll
  for (int i = 0; i < 4; ++i) { v[i] = __expf(v[i] - m); s += v[i]; }
  for (int o = 16; o > 0; o >>= 1) s += __shfl_xor(s, o, 32);
  const float inv = 1.0f / s;
#pragma unroll
  for (int i = 0; i < 4; ++i) {
    const float ww = v[i] * inv;
    w[(size_t)wave * Ss + lane + 32 * i] = ww;
    attn_out[((size_t)wave * Tt + t) * Ss + lane + 32 * i] = ww;
  }
}

// xcat[b, 0:H] = emb[token]
__global__ __launch_bounds__(512) void emb_gather(
    const int* __restrict__ tgt, const float* __restrict__ emb,
    float* __restrict__ xcat, int t) {
  const int b = blockIdx.x, h = threadIdx.x;
  const int tok = (t == 0) ? BOS_TOK : tgt[b * Tt + (t - 1)];
  xcat[(size_t)b * (2 * Hh) + h] = emb[(size_t)tok * Hh + h];
}

// xcat[b, H:2H] = sum_s w[b,s] * enc[b,s,:]
__global__ __launch_bounds__(512) void ctx_kernel(
    const float* __restrict__ w, const float* __restrict__ enc,
    float* __restrict__ xcat) {
  __shared__ float sw[Ss];
  const int b = blockIdx.x, h = threadIdx.x;
  if (threadIdx.x < Ss) sw[threadIdx.x] = w[(size_t)b * Ss + threadIdx.x];
  __syncthreads();
  const float* eb = enc + (size_t)b * Ss * Hh;
  float sum = 0.0f;
  for (int s = 0; s < Ss; ++s) sum += sw[s] * eb[(size_t)s * Hh + h];
  xcat[(size_t)b * (2 * Hh) + Hh + h] = sum;
}

// GRU gating elementwise
__global__ __launch_bounds__(256) void gru_gate(
    const float* __restrict__ gi, const float* __restrict__ gh,
    const float* __restrict__ h, float* __restrict__ hn) {
  const int idx = blockIdx.x * blockDim.x + threadIdx.x;
  if (idx >= Bb * Hh) return;
  const int b = idx / Hh, hh = idx - b * Hh;
  const float* gib = gi + (size_t)b * 3 * Hh;
  const float* ghb = gh + (size_t)b * 3 * Hh;
  const float r = 1.0f / (1.0f + __expf(-(gib[hh] + ghb[hh])));
  const float z = 1.0f / (1.0f + __expf(-(gib[Hh + hh] + ghb[Hh + hh])));
  const float n = tanhf(gib[2 * Hh + hh] + r * ghb[2 * Hh + hh]);
  hn[idx] = (1.0f - z) * n + z * h[idx];
}

// in-place log-softmax over V for each (b,t) row
__global__ __launch_bounds__(256) void logsoftmax_rows(float* __restrict__ x) {
  __shared__ float red[256];
  float* row = x + (size_t)blockIdx.x * Vv;
  const int tid = threadIdx.x;
  float m = -3.402823466e38f;
  for (int i = tid; i < Vv; i += 256) m = fmaxf(m, row[i]);
  red[tid] = m; __syncthreads();
  for (int o = 128; o > 0; o >>= 1) {
    if (tid < o) red[tid] = fmaxf(red[tid], red[tid + o]);
    __syncthreads();
  }
  m = red[0]; __syncthreads();
  float s = 0.0f;
  for (int i = tid; i < Vv; i += 256) s += __expf(row[i] - m);
  red[tid] = s; __syncthreads();
  for (int o = 128; o > 0; o >>= 1) {
    if (tid < o) red[tid] += red[tid + o];
    __syncthreads();
  }
  const float lse = m + __logf(red[0]);
  for (int i = tid; i < Vv; i += 256) row[i] = row[i] - lse;
}

__global__ __launch_bounds__(256) void copy_f32(
    const float* __restrict__ src, float* __restrict__ dst, int n) {
  const int i = blockIdx.x * blockDim.x + threadIdx.x;
  if (i < n) dst[i] = src[i];
}

// ---------------------------------------------------------------------------
extern "C" void kernel_launch(void* const* d_in, const int* in_sizes, int n_in,
                              void* d_out, int out_size, void* d_ws, size_t ws_size,
                              hipStream_t stream) {
  const float* enc   = (const float*)d_in[0];
  const float* h0    = (const float*)d_in[1];
  const int*   tgt   = (const int*)d_in[2];
  const float* emb   = (const float*)d_in[3];
  const float* Wa_w  = (const float*)d_in[4];
  const float* Wa_b  = (const float*)d_in[5];
  const float* Ua_w  = (const float*)d_in[6];
  const float* Ua_b  = (const float*)d_in[7];
  const float* Va_w  = (const float*)d_in[8];
  const float* Va_b  = (const float*)d_in[9];
  const float* W_ih  = (const float*)d_in[10];
  const float* b_ih  = (const float*)d_in[11];
  const float* W_hh  = (const float*)d_in[12];
  const float* b_hh  = (const float*)d_in[13];
  const float* out_w = (const float*)d_in[14];
  const float* out_b = (const float*)d_in[15];
  (void)in_sizes; (void)n_in; (void)out_size; (void)ws_size;

  float* out         = (float*)d_out;
  float* logits_base = out;                          // [B,T,V]
  float* hfinal_out  = out + (size_t)Bb * Tt * Vv;   // [1,B,H]
  float* attn_out    = hfinal_out + (size_t)Bb * Hh; // [B,T,S]

  float* ws = (float*)d_ws;
  size_t off = 0;
  float* ua   = ws + off; off += (size_t)Bb * Ss * Hh;   // 32 MB (L2-resident)
  float* q    = ws + off; off += (size_t)Bb * Hh;
  float* sc   = ws + off; off += (size_t)Bb * Ss;
  float* wat  = ws + off; off += (size_t)Bb * Ss;
  float* xcat = ws + off; off += (size_t)Bb * 2 * Hh;
  float* gi   = ws + off; off += (size_t)Bb * 3 * Hh;
  float* gh   = ws + off; off += (size_t)Bb * 3 * Hh;
  float* hA   = ws + off; off += (size_t)Bb * Hh;
  float* hB   = ws + off; off += (size_t)Bb * Hh;

  auto gemm = [&](const float* A, const float* W, const float* bias, float* C,
                  int M, int N, int K, int ldc) {
    const int waves  = (M / 64) * (N / 64);
    const int blocks = (waves + 7) / 8;   // 8 waves (256 thr) per block
    if (K == 512)
      wmma_gemm_nt<512><<<blocks, 256, 0, stream>>>(A, W, bias, C, M, N, ldc);
    else
      wmma_gemm_nt<1024><<<blocks, 256, 0, stream>>>(A, W, bias, C, M, N, ldc);
  };

  // Step-invariant: Ua_keys = enc @ Ua_w^T + Ua_b   [B*S, H]
  gemm(enc, Ua_w, Ua_b, ua, Bb * Ss, Hh, Hh, Hh);
  copy_f32<<<(Bb * Hh + 255) / 256, 256, 0, stream>>>(h0, hA, Bb * Hh);

  float* hbuf[2] = {hA, hB};
  for (int t = 0; t < Tt; ++t) {
    float* hc = hbuf[t & 1];
    float* hn = hbuf[(t + 1) & 1];
    // q = h @ Wa_w^T + Wa_b
    gemm(hc, Wa_w, Wa_b, q, Bb, Hh, Hh, Hh);
    attn_scores<<<(Bb * Ss + 7) / 8, 256, 0, stream>>>(q, ua, Va_w, Va_b, sc);
    attn_softmax<<<(Bb + 3) / 4, 128, 0, stream>>>(sc, wat, attn_out, t);
    emb_gather<<<Bb, Hh, 0, stream>>>(tgt, emb, xcat, t);
    ctx_kernel<<<Bb, Hh, 0, stream>>>(wat, enc, xcat);
    // gi = [x,ctx] @ W_ih^T + b_ih ; gh = h @ W_hh^T + b_hh
    gemm(xcat, W_ih, b_ih, gi, Bb, 3 * Hh, 2 * Hh, 3 * Hh);
    gemm(hc,   W_hh, b_hh, gh, Bb, 3 * Hh, Hh,     3 * Hh);
    gru_gate<<<(Bb * Hh + 255) / 256, 256, 0, stream>>>(gi, gh, hc, hn);
    // logits -> d_out[b, t, :]  (row stride T*V)
    gemm(hn, out_w, out_b, logits_base + (size_t)t * Vv, Bb, Vv, Hh, Tt * Vv);
  }

  logsoftmax_rows<<<Bb * Tt, 256, 0, stream>>>(logits_base);
  copy_f32<<<(Bb * Hh + 255) / 256, 256, 0, stream>>>(hbuf[Tt & 1], hfinal_out,
                                                      Bb * Hh);
}